// Mamba_59064390255104
// MI455X (gfx1250) — compile-verified
//
#include <hip/hip_runtime.h>
#include <math.h>

#define BSZ 8
#define TLEN 4096
#define DIM 2048
#define NC 16              // number of time chunks
#define LCH (TLEN / NC)    // 256 timesteps per chunk
#define CH 256             // channels per block (== blockDim.x)
#define TROWS 16           // timesteps staged per TDM tile
#define NSTAGE (LCH / TROWS)
#define LN_EPS 1e-5f

typedef unsigned int __attribute__((ext_vector_type(4))) u32x4;
typedef int          __attribute__((ext_vector_type(4))) i32x4;
typedef int          __attribute__((ext_vector_type(8))) i32x8;

#if defined(__gfx1250__) && __has_builtin(__builtin_amdgcn_tensor_load_to_lds) && __has_builtin(__builtin_amdgcn_s_wait_tensorcnt)
#define USE_TDM 1
#else
#define USE_TDM 0
#endif

#if USE_TDM
// Issue a TDM 2D tile load: rows x cols fp32 elements, row stride = stride
// elements in global memory, packed contiguously (cols*4 bytes/row) into LDS.
__device__ __forceinline__ void tdm_load_tile(const float* gsrc, unsigned lds_off,
                                              int rows, int cols, int stride) {
  unsigned long long ga = (unsigned long long)(uintptr_t)gsrc;
  u32x4 g0;
  g0.x = 1u;                                           // count=1, is_restore=0, gather=0
  g0.y = lds_off;                                      // lds_addr [63:32]
  g0.z = (unsigned)(ga & 0xFFFFFFFFu);                 // global_addr [95:64]
  g0.w = (unsigned)((ga >> 32) & 0x01FFFFFFu)          // global_addr [120:96]
       | (2u << 30);                                   // type=2 ("image")
  i32x8 g1;
  g1[0] = (2 << 16);                                   // data_size=2 (4 bytes)
  g1[1] = (cols & 0xFFFF) << 16;                       // tensor_dim0 lo16
  g1[2] = ((cols >> 16) & 0xFFFF) | ((rows & 0xFFFF) << 16); // dim0 hi | tensor_dim1 lo
  g1[3] = ((rows >> 16) & 0xFFFF) | ((cols & 0xFFFF) << 16); // dim1 hi | tile_dim0
  g1[4] = (rows & 0xFFFF);                             // tile_dim1 (tile_dim2=0)
  g1[5] = stride;                                      // tensor_dim0_stride lo32
  g1[6] = 0;                                           // stride0 hi | stride1 lo
  g1[7] = 0;                                           // stride1 hi
  i32x4 z4 = {0, 0, 0, 0};
  i32x8 z8 = {0, 0, 0, 0, 0, 0, 0, 0};
  __builtin_amdgcn_tensor_load_to_lds(g0, g1, z4, z4, z8, 0);
}
#endif

// ---------------------------------------------------------------------------
// Kernel 1: chunk-local scan. grid=(DIM/CH, NC, BSZ), block=CH threads.
// y_local written to yout; chunk-final local state to hend[c][b][d].
// ---------------------------------------------------------------------------
__global__ void __launch_bounds__(CH) ssm_scan_chunk(
    const float* __restrict__ x, const float* __restrict__ A,
    const float* __restrict__ Bp, const float* __restrict__ C,
    const float* __restrict__ Dp, float* __restrict__ yout,
    float* __restrict__ hend) {
  __shared__ float smem[2 * TROWS * CH];

  const int tid  = threadIdx.x;
  const int dblk = blockIdx.x;
  const int c    = blockIdx.y;
  const int b    = blockIdx.z;
  const int d    = dblk * CH + tid;

  const float a  = A[d];
  const float bp = Bp[d];
  const float cc = C[d];
  const float dp = Dp[d];

  const float* xbase = x    + ((size_t)b * TLEN + (size_t)c * LCH) * DIM + (size_t)dblk * CH;
  float*       ybase = yout + ((size_t)b * TLEN + (size_t)c * LCH) * DIM + (size_t)dblk * CH;

  float h = 0.0f;

#if USE_TDM
  const unsigned lds_base = (unsigned)(uintptr_t)(void*)smem;  // LDS byte offset
  if (tid < 32) {  // wave 0 drives the DMA engine
    tdm_load_tile(xbase, lds_base, TROWS, CH, DIM);
  }
  for (int s = 0; s < NSTAGE; ++s) {
    if (tid < 32) {
      if (s + 1 < NSTAGE) {
        tdm_load_tile(xbase + (size_t)(s + 1) * TROWS * DIM,
                      lds_base + (unsigned)(((s + 1) & 1) * TROWS * CH * 4),
                      TROWS, CH, DIM);
        __builtin_amdgcn_s_wait_tensorcnt(1);  // tile s done, s+1 in flight
      } else {
        __builtin_amdgcn_s_wait_tensorcnt(0);
      }
    }
    __syncthreads();
    const float* sb   = smem + (s & 1) * TROWS * CH;
    float*       yrow = ybase + (size_t)s * TROWS * DIM;
#pragma unroll
    for (int r = 0; r < TROWS; ++r) {
      float xv = sb[r * CH + tid];
      h = fmaf(a, h, bp * xv);
      yrow[(size_t)r * DIM + tid] = fmaf(cc, h, dp * xv);
    }
    __syncthreads();  // buffer (s&1) free for reuse at stage s+2
  }
#else
  for (int s = 0; s < NSTAGE; ++s) {
    float* sb = smem + (s & 1) * TROWS * CH;
    for (int i = tid; i < TROWS * CH; i += CH)
      sb[i] = xbase[(size_t)(s * TROWS + i / CH) * DIM + (i % CH)];
    __syncthreads();
    float* yrow = ybase + (size_t)s * TROWS * DIM;
#pragma unroll
    for (int r = 0; r < TROWS; ++r) {
      float xv = sb[r * CH + tid];
      h = fmaf(a, h, bp * xv);
      yrow[(size_t)r * DIM + tid] = fmaf(cc, h, dp * xv);
    }
    __syncthreads();
  }
#endif

  hend[((size_t)c * BSZ + b) * DIM + d] = h;
}

// ---------------------------------------------------------------------------
// Kernel 2: sequential chunk combine per (b,d): h_start chain with A^LCH.
// ---------------------------------------------------------------------------
__global__ void __launch_bounds__(256) ssm_combine(
    const float* __restrict__ A, const float* __restrict__ hend,
    float* __restrict__ hstart) {
  const int idx = blockIdx.x * 256 + threadIdx.x;  // < BSZ*DIM
  const int b = idx / DIM;
  const int d = idx % DIM;
  const float a = A[d];
  // A^LCH, LCH even -> positive; a==0 -> log2f(0)=-inf -> exp2f -> 0 (correct)
  const float apl = exp2f((float)LCH * log2f(fabsf(a)));
  float h = 0.0f;
#pragma unroll
  for (int c = 0; c < NC; ++c) {
    const size_t o = ((size_t)c * BSZ + b) * DIM + d;
    hstart[o] = h;
    h = fmaf(apl, h, hend[o]);
  }
}

// ---------------------------------------------------------------------------
// Kernel 3: cross-chunk correction + LayerNorm, in place on yout.
// grid=(TLEN, BSZ), block=256, 8 channels per thread.
// y_t += C * A^(j+1) * h_start[chunk];  then LN over D.
// ---------------------------------------------------------------------------
__global__ void __launch_bounds__(256) ssm_ln_fix(
    const float* __restrict__ A, const float* __restrict__ C,
    const float* __restrict__ hstart, const float* __restrict__ lnw,
    const float* __restrict__ lnb, float* __restrict__ y) {
  const int t   = blockIdx.x;
  const int b   = blockIdx.y;
  const int tid = threadIdx.x;
  const int c   = t / LCH;
  const int kk  = t - c * LCH + 1;  // power of A applied to h_start (1..LCH)

  const size_t rowbase = ((size_t)b * TLEN + t) * DIM;
  const float* hs = hstart + ((size_t)c * BSZ + b) * DIM;

  float vals[8];
  float sum = 0.0f, sumsq = 0.0f;
#pragma unroll
  for (int k = 0; k < 8; ++k) {
    const int d = tid + k * 256;
    const float a = A[d];
    float pk = exp2f((float)kk * log2f(fabsf(a)));  // |A|^kk (a==0 -> 0)
    if (a < 0.0f && (kk & 1)) pk = -pk;
    const float v = y[rowbase + d] + C[d] * pk * hs[d];
    vals[k] = v;
    sum += v;
    sumsq = fmaf(v, v, sumsq);
  }

  __shared__ float rs[256];
  __shared__ float rq[256];
  rs[tid] = sum;
  rq[tid] = sumsq;
  __syncthreads();
#pragma unroll
  for (int off = 128; off > 0; off >>= 1) {
    if (tid < off) {
      rs[tid] += rs[tid + off];
      rq[tid] += rq[tid + off];
    }
    __syncthreads();
  }
  const float mu  = rs[0] * (1.0f / DIM);
  const float var = rq[0] * (1.0f / DIM) - mu * mu;
  const float inv = rsqrtf(var + LN_EPS);

#pragma unroll
  for (int k = 0; k < 8; ++k) {
    const int d = tid + k * 256;
    y[rowbase + d] = (vals[k] - mu) * inv * lnw[d] + lnb[d];
  }
}

// ---------------------------------------------------------------------------
extern "C" void kernel_launch(void* const* d_in, const int* in_sizes, int n_in,
                              void* d_out, int out_size, void* d_ws, size_t ws_size,
                              hipStream_t stream) {
  const float* x   = (const float*)d_in[0];
  const float* A   = (const float*)d_in[1];
  const float* Bp  = (const float*)d_in[2];
  const float* C   = (const float*)d_in[3];
  const float* Dp  = (const float*)d_in[4];
  const float* lnw = (const float*)d_in[5];
  const float* lnb = (const float*)d_in[6];
  float* y = (float*)d_out;

  float* hend   = (float*)d_ws;                                  // NC*B*D floats (1 MB)
  float* hstart = (float*)d_ws + (size_t)NC * BSZ * DIM;         // NC*B*D floats (1 MB)

  dim3 g1(DIM / CH, NC, BSZ);  // 8 x 16 x 8 = 1024 blocks
  ssm_scan_chunk<<<g1, CH, 0, stream>>>(x, A, Bp, C, Dp, y, hend);

  ssm_combine<<<(BSZ * DIM) / 256, 256, 0, stream>>>(A, hend, hstart);

  dim3 g3(TLEN, BSZ);          // 32768 blocks
  ssm_ln_fix<<<g3, 256, 0, stream>>>(A, C, hstart, lnw, lnb, y);
}